// scaleEqNonlin1d_88459146428603
// MI455X (gfx1250) — compile-verified
//
#include <hip/hip_runtime.h>
#include <math.h>

#define B_    16
#define C_    128
#define N_    8192
#define NH    4096          // N/2
#define KBINS 4097          // N/2 + 1 rfft bins
#define KS    4104          // padded row stride (float2) for R workspace
#define WPAIR 64            // 128 K values -> 64 packed pairs per row

typedef __attribute__((ext_vector_type(2))) float v2f;
typedef __attribute__((ext_vector_type(8))) float v8f;

// ---------------------------------------------------------------------------
// Kernel 0: precompute the scaled DFT-128 twiddle tables, packed as the WMMA
// A-fragment K-pairs:  Wc[m][j] = scale*{cos(2pi m(2j)/128), cos(2pi m(2j+1)/128)}
//                      Ws[m][j] = scale*{sin(...), sin(...)}   (= -Im(W))
// 128 KB total -> lives in the 192 MB L2 for the whole of kernel 2.
// ---------------------------------------------------------------------------
__global__ __launch_bounds__(256) void k0_twiddle(float2* __restrict__ Wc,
                                                  float2* __restrict__ Ws)
{
    const int idx = blockIdx.x * 256 + threadIdx.x;   // m*64 + j
    const int m = idx >> 6, j = idx & 63;
    const float scale = 1.0f / 1048576.0f;            // 1/(C*N) = 2^-20
    // angle fraction for cospif: 2*pi*m*k/128 = pi * (m*k/64); exact in fp32
    float f0 = (float)(m * (2 * j))     * (1.f / 64.f);
    float f1 = (float)(m * (2 * j + 1)) * (1.f / 64.f);
    Wc[idx] = make_float2(cospif(f0) * scale, cospif(f1) * scale);
    Ws[idx] = make_float2(sinpif(f0) * scale, sinpif(f1) * scale);
}

// ---------------------------------------------------------------------------
// Kernel 1: one workgroup per (b,c) row.
//   LDS holds the full 8192-pt complex line (64 KB of the 320 KB WGP LDS).
//   DIF inverse FFT (natural in -> bit-reversed out), take Re, instance-norm
//   + exact GELU (order-invariant under the bit-reversal permutation), then
//   DIT forward FFT (bit-reversed in -> natural out): bit-reversal is free.
//   Writes rfft bins 0..4096 to workspace R[b][c][k] (interleaved float2).
// ---------------------------------------------------------------------------
__global__ __launch_bounds__(256) void k1_row_fft(const float* __restrict__ xre,
                                                  const float* __restrict__ xim,
                                                  const float* __restrict__ gamma,
                                                  const float* __restrict__ beta,
                                                  float2* __restrict__ R)
{
    __shared__ float re[N_];
    __shared__ float im[N_];
    const int row = blockIdx.x;            // b*C + c
    const int c   = row & (C_ - 1);
    const int tid = threadIdx.x;
    const size_t base = (size_t)row * N_;

    for (int i = tid; i < N_; i += 256) {
        re[i] = xre[base + i];
        im[i] = xim[base + i];
    }
    __syncthreads();

    // Unscaled inverse FFT, DIF radix-2: natural input -> bit-reversed output
    for (int ls = 12; ls >= 0; --ls) {
        const int s = 1 << ls;
        for (int t = tid; t < NH; t += 256) {
            const int j  = t & (s - 1);
            const int i0 = ((t >> ls) << (ls + 1)) + j;
            const int i1 = i0 + s;
            float ar = re[i0], ai = im[i0];
            float br = re[i1], bi = im[i1];
            float a  = (float)j / (float)s;       // exact: s is a power of 2
            float wr = cospif(a);
            float wi = sinpif(a);                 // + sign: inverse transform
            float tr = ar - br, ti = ai - bi;
            re[i0] = ar + br;  im[i0] = ai + bi;
            re[i1] = tr * wr - ti * wi;
            im[i1] = tr * wi + ti * wr;
        }
        __syncthreads();
    }

    // Instance-norm stats over Re(x): sums are permutation-invariant.
    float lsum = 0.f, lsq = 0.f;
    for (int i = tid; i < N_; i += 256) { float v = re[i]; lsum += v; lsq += v * v; }
    im[tid]       = lsum;        // im[] holds the (discarded) imag part: reuse
    im[256 + tid] = lsq;
    __syncthreads();
    for (int off = 128; off > 0; off >>= 1) {
        if (tid < off) { im[tid] += im[tid + off]; im[256 + tid] += im[256 + tid + off]; }
        __syncthreads();
    }
    const float mean = im[0]   * (1.f / N_);
    const float var  = im[256] * (1.f / N_) - mean * mean;
    const float istd = rsqrtf(var + 1e-5f);
    const float gm = gamma[c], bt = beta[c];
    __syncthreads();

    // Normalize + exact GELU; zero imag for the forward real FFT
    for (int i = tid; i < N_; i += 256) {
        float v = (re[i] - mean) * istd * gm + bt;
        v = 0.5f * v * (1.f + erff(v * 0.70710678118654752f));
        re[i] = v;
        im[i] = 0.f;
    }
    __syncthreads();

    // Unscaled forward FFT, DIT radix-2: bit-reversed input -> natural output
    for (int ls = 0; ls <= 12; ++ls) {
        const int s = 1 << ls;
        for (int t = tid; t < NH; t += 256) {
            const int j  = t & (s - 1);
            const int i0 = ((t >> ls) << (ls + 1)) + j;
            const int i1 = i0 + s;
            float a  = (float)j / (float)s;
            float wr = cospif(a);
            float wi = -sinpif(a);                // - sign: forward transform
            float br = re[i1], bi = im[i1];
            float tr = br * wr - bi * wi;
            float ti = br * wi + bi * wr;
            float ar = re[i0], ai = im[i0];
            re[i0] = ar + tr;  im[i0] = ai + ti;
            re[i1] = ar - tr;  im[i1] = ai - ti;
        }
        __syncthreads();
    }

    float2* Rrow = R + (size_t)row * KS;
    for (int k = tid; k < KBINS; k += 256)
        Rrow[k] = make_float2(re[k], im[k]);
}

// ---------------------------------------------------------------------------
// Kernel 2: size-128 DFT over the channel axis as a complex GEMM with fp32
// WMMA:  Out[b][:, k] = W(128x128) * R[b][:, k].
// Each block: one b, 32 k-columns staged in LDS (row stride 40 -> the two
// lane groups of the B fragment hit disjoint banks; the two 16-col tiles hit
// disjoint bank quarters). 8 waves cover M=128; single K-loop of 32 steps
// keeps 4 accumulators (re/im x 2 col tiles) and does 8 V_WMMA_F32_16X16X4_F32
// per step with A fragments loaded straight from the L2-resident tables.
// Epilogue writes bin k and its Hermitian mirror N-k (Im dropped at k=0, N/2).
// ---------------------------------------------------------------------------
__global__ __launch_bounds__(256) void k2_dftC_wmma(const float2* __restrict__ R,
                                                    const float2* __restrict__ Wc,
                                                    const float2* __restrict__ Ws,
                                                    float* __restrict__ outre,
                                                    float* __restrict__ outim)
{
    __shared__ float Lre[C_][40];
    __shared__ float Lim[C_][40];
    const int b    = blockIdx.y;
    const int col0 = blockIdx.x * 32;
    const int tid  = threadIdx.x;
    const float2* Rb = R + (size_t)b * C_ * KS;

    for (int idx = tid; idx < C_ * 32; idx += 256) {
        int r = idx >> 5, cc = idx & 31;
        int k = col0 + cc;
        float2 v = (k < KBINS) ? Rb[(size_t)r * KS + k] : make_float2(0.f, 0.f);
        Lre[r][cc] = v.x;
        Lim[r][cc] = v.y;
    }
    __syncthreads();

    const int wave = tid >> 5;          // 0..7 -> M tile (rows 16*wave..+15)
    const int lane = tid & 31;
    const int lg   = lane >> 4;         // lane group: K split for A/B fragments
    const int ln   = lane & 15;
    const int m    = wave * 16 + ln;    // A-matrix row held by this lane
    const float2* WcRow = Wc + m * WPAIR;
    const float2* WsRow = Ws + m * WPAIR;

    v8f accr0 = {0.f,0.f,0.f,0.f,0.f,0.f,0.f,0.f};
    v8f acci0 = {0.f,0.f,0.f,0.f,0.f,0.f,0.f,0.f};
    v8f accr1 = {0.f,0.f,0.f,0.f,0.f,0.f,0.f,0.f};
    v8f acci1 = {0.f,0.f,0.f,0.f,0.f,0.f,0.f,0.f};

    for (int k0 = 0; k0 < C_; k0 += 4) {
        const int kA = k0 + 2 * lg;           // this lane's A/B K-pair
        // A fragments from L2-resident tables (one b64 load each)
        float2 wc = WcRow[(k0 >> 1) + lg];
        float2 wsn = WsRow[(k0 >> 1) + lg];
        v2f aRe = {  wc.x,  wc.y  };          //  Re(W)*scale
        v2f aPs = {  wsn.x, wsn.y };          // -Im(W)*scale (= +sin)
        v2f aNs = { -wsn.x, -wsn.y };         //  Im(W)*scale
        // B fragments for the two 16-column tiles (conflict-free LDS reads)
        v2f bRe0 = { Lre[kA][ln],      Lre[kA + 1][ln]      };
        v2f bIm0 = { Lim[kA][ln],      Lim[kA + 1][ln]      };
        v2f bRe1 = { Lre[kA][16 + ln], Lre[kA + 1][16 + ln] };
        v2f bIm1 = { Lim[kA][16 + ln], Lim[kA + 1][16 + ln] };
        // complex MAC: out_re += ReW*ReR - ImW*ImR ; out_im += ReW*ImR + ImW*ReR
        accr0 = __builtin_amdgcn_wmma_f32_16x16x4_f32(false, aRe, false, bRe0, (short)0, accr0, false, false);
        accr0 = __builtin_amdgcn_wmma_f32_16x16x4_f32(false, aPs, false, bIm0, (short)0, accr0, false, false);
        acci0 = __builtin_amdgcn_wmma_f32_16x16x4_f32(false, aRe, false, bIm0, (short)0, acci0, false, false);
        acci0 = __builtin_amdgcn_wmma_f32_16x16x4_f32(false, aNs, false, bRe0, (short)0, acci0, false, false);
        accr1 = __builtin_amdgcn_wmma_f32_16x16x4_f32(false, aRe, false, bRe1, (short)0, accr1, false, false);
        accr1 = __builtin_amdgcn_wmma_f32_16x16x4_f32(false, aPs, false, bIm1, (short)0, accr1, false, false);
        acci1 = __builtin_amdgcn_wmma_f32_16x16x4_f32(false, aRe, false, bIm1, (short)0, acci1, false, false);
        acci1 = __builtin_amdgcn_wmma_f32_16x16x4_f32(false, aNs, false, bRe1, (short)0, acci1, false, false);
    }

    for (int ct = 0; ct < 2; ++ct) {
        const v8f& ar = ct ? accr1 : accr0;
        const v8f& ai = ct ? acci1 : acci0;
        const int kcol = col0 + ct * 16 + ln;
        if (kcol < KBINS) {
            const bool edge = (kcol == 0) || (kcol == NH);  // Im dropped by irfft
            for (int v = 0; v < 8; ++v) {
                const int mr = wave * 16 + 8 * lg + v;      // C/D row layout
                const size_t o = (size_t)(b * C_ + mr) * N_;
                float orv = ar[v];
                float oiv = edge ? 0.f : ai[v];
                outre[o + kcol] = orv;
                outim[o + kcol] = oiv;
                if (kcol >= 1 && kcol <= NH - 1) {          // Hermitian mirror
                    outre[o + (N_ - kcol)] =  orv;
                    outim[o + (N_ - kcol)] = -oiv;
                }
            }
        }
    }
}

extern "C" void kernel_launch(void* const* d_in, const int* in_sizes, int n_in,
                              void* d_out, int out_size, void* d_ws, size_t ws_size,
                              hipStream_t stream) {
    const float* xre   = (const float*)d_in[0];
    const float* xim   = (const float*)d_in[1];
    const float* gamma = (const float*)d_in[2];
    const float* beta  = (const float*)d_in[3];
    float* outre = (float*)d_out;
    float* outim = outre + (size_t)B_ * C_ * N_;

    // workspace layout: R tiles, then the two twiddle tables (16B aligned)
    float2* R  = (float2*)d_ws;                       // B*C*KS float2 = ~64.1 MiB
    float2* Wc = R + (size_t)B_ * C_ * KS;            // 8192 float2 = 64 KiB
    float2* Ws = Wc + C_ * WPAIR;                     // 8192 float2 = 64 KiB

    k0_twiddle<<<dim3((C_ * WPAIR) / 256), dim3(256), 0, stream>>>(Wc, Ws);
    k1_row_fft<<<dim3(B_ * C_), dim3(256), 0, stream>>>(xre, xim, gamma, beta, R);
    k2_dftC_wmma<<<dim3((KBINS + 31) / 32, B_), dim3(256), 0, stream>>>(R, Wc, Ws, outre, outim);
}